// Encoder_11433202942735
// MI455X (gfx1250) — compile-verified
//
#include <hip/hip_runtime.h>
#include <hip/hip_bf16.h>

#define N_NODES 100000
#define N_EDGES 3200000
#define IN_DIM  512
#define OUT_DIM 256

typedef __attribute__((ext_vector_type(2))) float v2f;
typedef __attribute__((ext_vector_type(4))) float v4f;
typedef __attribute__((ext_vector_type(8))) float v8f;

// ---------------------------------------------------------------------------
// Kernel 1: zero the mean-accumulator half of d_out (N*256 f32) and deg[N].
// ---------------------------------------------------------------------------
__global__ __launch_bounds__(256)
void k_zero(v4f* __restrict__ sums4, float* __restrict__ deg) {
  unsigned i = blockIdx.x * 256u + threadIdx.x;
  const unsigned tot4 = (unsigned)N_NODES * (OUT_DIM / 4);  // 6.4M float4
  if (i < tot4) sums4[i] = (v4f){0.f, 0.f, 0.f, 0.f};
  if (i < (unsigned)N_NODES) deg[i] = 0.f;
}

// ---------------------------------------------------------------------------
// Kernel 2: z = L2normalize(relu(h @ W^T + b)) via V_WMMA_F32_16X16X4_F32.
// One block = 16 waves = one 16-row M tile; wave w owns N columns [16w,16w+16).
// Per-lane fragment layout per CDNA5 ISA 7.12.2 (wave32):
//   A (16x4 f32):  a[v] = h[mtile*16 + lane%16][k + v + 2*(lane/16)]
//   B (4x16 f32):  b[v] = W[wave*16 + lane%16][k + v + 2*(lane/16)]   (B=W^T)
//   C/D (16x16):   c[v] = z[row = v + 8*(lane/16)][col = lane%16]
// ---------------------------------------------------------------------------
__global__ __launch_bounds__(512)
void k_gemm_relu_l2(const float* __restrict__ h, const float* __restrict__ W,
                    const float* __restrict__ bias, float* __restrict__ z) {
  __shared__ float s_ssq[16];

  const int lane  = threadIdx.x & 31;
  const int wave  = threadIdx.x >> 5;   // n-tile 0..15
  const int mtile = blockIdx.x;         // 0..6249 (100000 = 16*6250, no tail)
  const int l16   = lane & 15;
  const int hi    = lane >> 4;          // 0: lanes 0-15, 1: lanes 16-31
  const int mrow  = mtile * 16 + l16;
  const int ncol  = wave * 16 + l16;

  const float* __restrict__ arow = h + (size_t)mrow * IN_DIM + 2 * hi;
  const float* __restrict__ brow = W + (size_t)ncol * IN_DIM + 2 * hi;

  if (threadIdx.x < 16) s_ssq[threadIdx.x] = 0.f;

  v8f acc = {0.f, 0.f, 0.f, 0.f, 0.f, 0.f, 0.f, 0.f};
#pragma unroll 8
  for (int k = 0; k < IN_DIM; k += 4) {
    v2f a = *(const v2f*)(arow + k);
    v2f b = *(const v2f*)(brow + k);
    // 8 args: (neg_a, A, neg_b, B, c_mod, C, reuse_a, reuse_b)
    acc = __builtin_amdgcn_wmma_f32_16x16x4_f32(false, a, false, b,
                                                (short)0, acc, false, false);
  }

  // bias + ReLU, per-lane partial sum-of-squares
  const float bn = bias[ncol];
  float sq[8];
#pragma unroll
  for (int v = 0; v < 8; ++v) {
    float t = acc[v] + bn;
    t = t > 0.f ? t : 0.f;
    acc[v] = t;
    sq[v]  = t * t;
  }
  // reduce across the 16 columns held by each 16-lane half (wave32 butterfly)
#pragma unroll
  for (int off = 1; off < 16; off <<= 1) {
#pragma unroll
    for (int v = 0; v < 8; ++v) sq[v] += __shfl_xor(sq[v], off, 32);
  }
  __syncthreads();  // s_ssq zero-init visible to all 16 waves
  if (l16 == 0) {
#pragma unroll
    for (int v = 0; v < 8; ++v) atomicAdd(&s_ssq[v + 8 * hi], sq[v]);  // ds_add_f32
  }
  __syncthreads();

#pragma unroll
  for (int v = 0; v < 8; ++v) {
    const int row = v + 8 * hi;
    const float inv = 1.0f / fmaxf(sqrtf(s_ssq[row]), 1e-12f);
    z[(size_t)(mtile * 16 + row) * OUT_DIM + ncol] = acc[v] * inv;
  }
}

// ---------------------------------------------------------------------------
// Kernel 3: one wave32 per edge; coalesced gather of z[src] row + coalesced
// global_atomic_add_f32 scatter into sums[dst]. Lane 0 counts the degree.
// z (102 MB) + sums (102 MB) both live in the 192 MB L2 -> atomics stay on-chip.
// ---------------------------------------------------------------------------
__global__ __launch_bounds__(256)
void k_scatter(const float* __restrict__ z, const int* __restrict__ src,
               const int* __restrict__ dst, float* __restrict__ sums,
               float* __restrict__ deg) {
  const int lane = threadIdx.x & 31;
  const int e = (int)((blockIdx.x * 256u + threadIdx.x) >> 5);
  if (e >= N_EDGES) return;
  const int s = src[e];
  const int d = dst[e];
  if (lane == 0) atomicAdd(&deg[d], 1.0f);
  const float* __restrict__ zs  = z    + (size_t)s * OUT_DIM + lane;
  float*       __restrict__ out = sums + (size_t)d * OUT_DIM + lane;
#pragma unroll
  for (int k = 0; k < OUT_DIM; k += 32) atomicAdd(out + k, zs[k]);
}

// ---------------------------------------------------------------------------
// Kernel 4: mean = sums / deg (deg>0), else 0. float4-vectorized, in place.
// ---------------------------------------------------------------------------
__global__ __launch_bounds__(256)
void k_finalize(v4f* __restrict__ mean4, const float* __restrict__ deg) {
  unsigned i = blockIdx.x * 256u + threadIdx.x;
  const unsigned tot4 = (unsigned)N_NODES * (OUT_DIM / 4);
  if (i >= tot4) return;
  const float dg  = deg[i >> 6];            // 64 float4 chunks per node
  const float inv = dg > 0.f ? 1.0f / dg : 0.f;
  v4f v = mean4[i];
  v *= inv;
  mean4[i] = v;
}

// ---------------------------------------------------------------------------
extern "C" void kernel_launch(void* const* d_in, const int* in_sizes, int n_in,
                              void* d_out, int out_size, void* d_ws, size_t ws_size,
                              hipStream_t stream) {
  const float* h   = (const float*)d_in[0];   // [100000, 512]
  const float* W   = (const float*)d_in[1];   // [256, 512]
  const float* b   = (const float*)d_in[2];   // [256]
  const int*   src = (const int*)d_in[3];     // [3.2M]
  const int*   dst = (const int*)d_in[4];     // [3.2M]

  float* z    = (float*)d_out;                       // output 0: [100000, 256]
  float* mean = z + (size_t)N_NODES * OUT_DIM;       // output 1: [100000, 256]
  float* deg  = (float*)d_ws;                        // [100000] f32 scratch

  const int tot4 = N_NODES * (OUT_DIM / 4);          // 6,400,000

  k_zero<<<(tot4 + 255) / 256, 256, 0, stream>>>((v4f*)mean, deg);
  k_gemm_relu_l2<<<N_NODES / 16, 512, 0, stream>>>(h, W, b, z);
  k_scatter<<<(N_EDGES * 32 + 255) / 256, 256, 0, stream>>>(z, src, dst, mean, deg);
  k_finalize<<<(tot4 + 255) / 256, 256, 0, stream>>>((v4f*)mean, deg);
}